// XY2UV_41970420417781
// MI455X (gfx1250) — compile-verified
//
#include <hip/hip_runtime.h>
#include <hip/hip_bf16.h>
#include <stdint.h>

#define BB 8
#define CC 3
#define HH 512
#define WW 512
#define VV 10475
#define FF 20908
#define UHD 512
#define UWD 512
#define UHW (UHD * UWD)

// ---------------- CDNA5 async global->LDS helpers (gfx1250) ----------------

#if defined(__HIP_DEVICE_COMPILE__)
typedef __attribute__((address_space(1))) int gas_int;
typedef __attribute__((address_space(3))) int las_int;
#endif

__device__ __forceinline__ void cdna5_async_load_b32(const void* gsrc, void* lds_dst) {
#if defined(__HIP_DEVICE_COMPILE__)
#if __has_builtin(__builtin_amdgcn_global_load_async_to_lds_b32)
  __builtin_amdgcn_global_load_async_to_lds_b32(
      (gas_int*)gsrc, (las_int*)lds_dst, /*offset=*/0, /*cpol=*/0);
#else
  unsigned loff = (unsigned)(uintptr_t)(__attribute__((address_space(3))) char*)lds_dst;
  asm volatile("global_load_async_to_lds_b32 %0, %1, off"
               :: "v"(loff), "v"((unsigned long long)(uintptr_t)gsrc)
               : "memory");
#endif
#else
  (void)gsrc;
  (void)lds_dst;
#endif
}

__device__ __forceinline__ void cdna5_wait_async0() {
#if defined(__HIP_DEVICE_COMPILE__)
#if __has_builtin(__builtin_amdgcn_s_wait_asynccnt)
  __builtin_amdgcn_s_wait_asynccnt(0);
#else
  asm volatile("s_wait_asynccnt 0" ::: "memory");
#endif
#endif
}

// ---------------- Kernel 1: perspective-project mesh vertices --------------

__global__ __launch_bounds__(256) void k_project(const float* __restrict__ mc,
                                                 const float* __restrict__ focal,
                                                 const float* __restrict__ princpt,
                                                 float* __restrict__ mimg) {
  int i = blockIdx.x * 256 + threadIdx.x;
  if (i >= BB * VV) return;
  int b = i / VV;
  float x = mc[3 * i + 0];
  float y = mc[3 * i + 1];
  float z = mc[3 * i + 2];
  mimg[2 * i + 0] = x / z * focal[2 * b + 0] + princpt[2 * b + 0];
  mimg[2 * i + 1] = y / z * focal[2 * b + 1] + princpt[2 * b + 1];
}

// ---------------- Kernel 2: clear visibility mask --------------------------

__global__ __launch_bounds__(256) void k_clear(unsigned char* __restrict__ v, int n) {
  int i = blockIdx.x * 256 + threadIdx.x;
  if (i < n) v[i] = 0;
}

// ---------------- Kernel 3: scatter visibility from pix_to_face_xy ---------

__global__ __launch_bounds__(256) void k_scatter(const int* __restrict__ p2fxy,
                                                 unsigned char* __restrict__ v) {
  int i = blockIdx.x * 256 + threadIdx.x;  // over B*H*W = 2,097,152
  if (i >= BB * HH * WW) return;
  int b = i / (HH * WW);
  int p = p2fxy[i];
  int local = (p >= 0) ? (p - b * FF) : (FF - 1);  // matches reference quirk
  v[b * FF + local] = 1;                            // benign race: all write 1
}

// ---------------- Kernel 4: main uv-map kernel -----------------------------

__global__ __launch_bounds__(256) void k_main(const float* __restrict__ img,
                                              const int* __restrict__ face,
                                              const int* __restrict__ p2fuv,
                                              const float* __restrict__ bary,
                                              const float* __restrict__ mimg,
                                              const unsigned char* __restrict__ validb,
                                              float* __restrict__ out) {
  __shared__ int sh_p2f[256];
  __shared__ float sh_bary[768];

  const int t = threadIdx.x;
  const int gbase = blockIdx.x * 256;  // in [0, B*UHW); 256 | UHW so single b/block
  const int b = gbase / UHW;
  const int p0 = gbase - b * UHW;

  // Async-stage the streaming operands into LDS (ASYNCcnt path).
  cdna5_async_load_b32(p2fuv + p0 + t, &sh_p2f[t]);
  const float* bb = bary + (size_t)p0 * 3;
  cdna5_async_load_b32(bb + t, &sh_bary[t]);
  cdna5_async_load_b32(bb + t + 256, &sh_bary[t + 256]);
  cdna5_async_load_b32(bb + t + 512, &sh_bary[t + 512]);
  cdna5_wait_async0();
  __syncthreads();

  const int p = p0 + t;
  const int g = sh_p2f[t];
  const int gi = (g < 0) ? (FF + g) : g;  // -1 -> F-1
  const bool vis = validb[b * FF + gi] != 0;

  float r0, r1, r2;
  if (!vis || g == -1) {
    r0 = r1 = r2 = -1.0f;
  } else {
    const int f0 = face[3 * gi + 0];
    const int f1 = face[3 * gi + 1];
    const int f2 = face[3 * gi + 2];
    const float w0 = sh_bary[3 * t + 0];
    const float w1 = sh_bary[3 * t + 1];
    const float w2 = sh_bary[3 * t + 2];
    const float* q0 = mimg + 2 * ((size_t)b * VV + f0);
    const float* q1 = mimg + 2 * ((size_t)b * VV + f1);
    const float* q2 = mimg + 2 * ((size_t)b * VV + f2);
    float bx = q0[0] * w0 + q1[0] * w1 + q2[0] * w2;
    float by = q0[1] * w0 + q1[1] * w1 + q2[1] * w2;

    // normalize -> denormalize exactly as reference does
    float gridx = bx / (float)(WW - 1) * 2.0f - 1.0f;
    float gridy = by / (float)(HH - 1) * 2.0f - 1.0f;
    float gx = (gridx + 1.0f) * 0.5f * (float)(WW - 1);
    float gy = (gridy + 1.0f) * 0.5f * (float)(HH - 1);

    float x0 = floorf(gx), y0 = floorf(gy);
    float wx = gx - x0, wy = gy - y0;
    float x1 = x0 + 1.0f, y1 = y0 + 1.0f;
    float w00 = (1.0f - wx) * (1.0f - wy);
    float w10 = wx * (1.0f - wy);
    float w01 = (1.0f - wx) * wy;
    float w11 = wx * wy;

    bool okx0 = (x0 >= 0.0f) && (x0 <= (float)(WW - 1));
    bool okx1 = (x1 >= 0.0f) && (x1 <= (float)(WW - 1));
    bool oky0 = (y0 >= 0.0f) && (y0 <= (float)(HH - 1));
    bool oky1 = (y1 >= 0.0f) && (y1 <= (float)(HH - 1));
    int ix0 = (int)fminf(fmaxf(x0, 0.0f), (float)(WW - 1));
    int ix1 = (int)fminf(fmaxf(x1, 0.0f), (float)(WW - 1));
    int iy0 = (int)fminf(fmaxf(y0, 0.0f), (float)(HH - 1));
    int iy1 = (int)fminf(fmaxf(y1, 0.0f), (float)(HH - 1));

    const size_t chan = (size_t)HH * WW;
    const float* ib = img + (size_t)b * CC * chan;
    int o00 = iy0 * WW + ix0;
    int o10 = iy0 * WW + ix1;
    int o01 = iy1 * WW + ix0;
    int o11 = iy1 * WW + ix1;
    bool ok00 = okx0 && oky0, ok10 = okx1 && oky0;
    bool ok01 = okx0 && oky1, ok11 = okx1 && oky1;

    float res[3];
#pragma unroll
    for (int c = 0; c < 3; ++c) {
      const float* pc = ib + (size_t)c * chan;
      float t00 = ok00 ? pc[o00] : 0.0f;
      float t10 = ok10 ? pc[o10] : 0.0f;
      float t01 = ok01 ? pc[o01] : 0.0f;
      float t11 = ok11 ? pc[o11] : 0.0f;
      res[c] = t00 * w00 + t10 * w10 + t01 * w01 + t11 * w11;
    }
    r0 = res[0];
    r1 = res[1];
    r2 = res[2];
  }

  // mask = (v != -1); uvmap = v * mask   (outputs concatenated: uvmap, mask)
  const size_t half = (size_t)BB * CC * UHW;
  float m0 = (r0 != -1.0f) ? 1.0f : 0.0f;
  float m1 = (r1 != -1.0f) ? 1.0f : 0.0f;
  float m2 = (r2 != -1.0f) ? 1.0f : 0.0f;
  size_t o0 = (size_t)(b * CC + 0) * UHW + p;
  size_t o1 = (size_t)(b * CC + 1) * UHW + p;
  size_t o2 = (size_t)(b * CC + 2) * UHW + p;
  out[o0] = r0 * m0;
  out[o1] = r1 * m1;
  out[o2] = r2 * m2;
  out[half + o0] = m0;
  out[half + o1] = m1;
  out[half + o2] = m2;
}

// ---------------- Host launch ----------------------------------------------

extern "C" void kernel_launch(void* const* d_in, const int* in_sizes, int n_in,
                              void* d_out, int out_size, void* d_ws, size_t ws_size,
                              hipStream_t stream) {
  const float* img = (const float*)d_in[0];
  const float* mesh_cam = (const float*)d_in[1];
  const int* face = (const int*)d_in[2];
  const float* focal = (const float*)d_in[3];
  const float* princpt = (const float*)d_in[4];
  const int* p2fxy = (const int*)d_in[5];
  const int* p2fuv = (const int*)d_in[6];
  const float* bary = (const float*)d_in[7];
  float* out = (float*)d_out;

  float* mimg = (float*)d_ws;  // B*V*2 floats = 670,400 B (16-aligned)
  unsigned char* validb = (unsigned char*)d_ws + (size_t)BB * VV * 2 * sizeof(float);

  k_project<<<(BB * VV + 255) / 256, 256, 0, stream>>>(mesh_cam, focal, princpt, mimg);
  k_clear<<<(BB * FF + 255) / 256, 256, 0, stream>>>(validb, BB * FF);
  k_scatter<<<(BB * HH * WW + 255) / 256, 256, 0, stream>>>(p2fxy, validb);
  k_main<<<(BB * UHW) / 256, 256, 0, stream>>>(img, face, p2fuv, bary, mimg, validb, out);
}